// GraphSAGEBlock_66932770341396
// MI455X (gfx1250) — compile-verified
//
#include <hip/hip_runtime.h>

typedef __attribute__((ext_vector_type(2))) float v2f;
typedef __attribute__((ext_vector_type(8))) float v8f;

#define FD 96            // feature dim (fixed by the problem)
#define BN_EPS 1e-5f

// ---------------------------------------------------------------------------
// Zero a float region (graph-capture-safe replacement for hipMemsetAsync).
// ---------------------------------------------------------------------------
__global__ void zero_f32(float* __restrict__ p, int n) {
    int i = blockIdx.x * blockDim.x + threadIdx.x;
    if (i < n) p[i] = 0.0f;
}

// ---------------------------------------------------------------------------
// Scatter: one wave32 per edge. Lanes cooperatively atomic-add the 96-float
// source row into msg[dst]; lane 0 bumps the degree counter.
// Emits global_atomic_add_f32 without return (STOREcnt path).
// ---------------------------------------------------------------------------
__global__ __launch_bounds__(256) void sage_scatter(
    const float* __restrict__ x,
    const int*   __restrict__ ei,   // [2, E] (src row, then dst row)
    float* __restrict__ msg,        // [N, 96]
    float* __restrict__ deg,        // [N]
    int E)
{
    int wave = (blockIdx.x * blockDim.x + threadIdx.x) >> 5;
    int lane = threadIdx.x & 31;
    if (wave >= E) return;
    int s = ei[wave];
    int d = ei[E + wave];
    const float* xr = x + (long long)s * FD;
    float* mr = msg + (long long)d * FD;
#pragma unroll
    for (int j = 0; j < 3; ++j) {
        int f = lane + 32 * j;
        atomicAdd(&mr[f], xr[f]);
    }
    if (lane == 0) atomicAdd(&deg[d], 1.0f);
}

// ---------------------------------------------------------------------------
// Fused dual-GEMM with V_WMMA_F32_16X16X4_F32:
//   h = (msg/deg) @ Wl^T + bl + x @ Wr^T
// One wave per 16x16 output tile; block of 6 waves covers all 96 columns of a
// 16-row node tile. Also accumulates per-column sum / sum-of-squares for BN.
// Requires N % 16 == 0 (true: 50000 = 3125*16) so EXEC is all-1s for WMMA.
// ---------------------------------------------------------------------------
__global__ __launch_bounds__(192) void sage_gemm_wmma(
    const float* __restrict__ x,
    const float* __restrict__ msg,
    const float* __restrict__ deg,
    const float* __restrict__ Wl,   // [96,96] row-major: Wl[o][d]
    const float* __restrict__ bl,   // [96]
    const float* __restrict__ Wr,   // [96,96]
    float* __restrict__ h,          // [N,96]  (d_out reused as scratch)
    float* __restrict__ colsum,     // [96]
    float* __restrict__ colsumsq)   // [96]
{
    const int lane   = threadIdx.x & 31;
    const int wv     = threadIdx.x >> 5;      // 0..5 -> column tile
    const int m_base = blockIdx.x << 4;       // 16 rows per block
    const int n_base = wv << 4;

    const int m   = lane & 15;                // A row held by this lane
    const int kA  = (lane >> 4) << 1;         // K sub-offset: 0 or 2
    const int row = m_base + m;
    const int col = n_base + (lane & 15);     // B/C/D column held by this lane

    const float invdeg = 1.0f / fmaxf(deg[row], 1.0f);

    const float* __restrict__ Arow = msg + (long long)row * FD;
    const float* __restrict__ Xrow = x   + (long long)row * FD;
    const float* __restrict__ Brl  = Wl  + (long long)col * FD;  // B[k][col]=Wl[col][k]
    const float* __restrict__ Brr  = Wr  + (long long)col * FD;

    v8f c = {};

    // GEMM 1: neigh_mean @ Wl^T   (fold 1/deg into the A fragment)
#pragma unroll
    for (int kk = 0; kk < FD; kk += 4) {
        v2f a = *(const v2f*)(Arow + kk + kA);
        a.x *= invdeg;
        a.y *= invdeg;
        v2f b = *(const v2f*)(Brl + kk + kA);
        c = __builtin_amdgcn_wmma_f32_16x16x4_f32(
                false, a, false, b, (short)0, c, false, false);
    }
    // GEMM 2: x @ Wr^T, accumulating into the same C
#pragma unroll
    for (int kk = 0; kk < FD; kk += 4) {
        v2f a = *(const v2f*)(Xrow + kk + kA);
        v2f b = *(const v2f*)(Brr + kk + kA);
        c = __builtin_amdgcn_wmma_f32_16x16x4_f32(
                false, a, false, b, (short)0, c, false, false);
    }

    // Epilogue: + bias, store h, accumulate BN statistics.
    const float bias = bl[col];
    const int   hl   = lane >> 4;             // which half of the M rows
    float s1 = 0.0f, s2 = 0.0f;
#pragma unroll
    for (int r = 0; r < 8; ++r) {
        float v = c[r] + bias;
        h[(long long)(m_base + r + 8 * hl) * FD + col] = v;
        s1 += v;
        s2 += v * v;
    }
    // Combine the two M-halves of this column (lanes l and l^16).
    s1 += __shfl_xor(s1, 16, 32);
    s2 += __shfl_xor(s2, 16, 32);
    if (lane < 16) {
        atomicAdd(&colsum[col],   s1);
        atomicAdd(&colsumsq[col], s2);
    }
}

// ---------------------------------------------------------------------------
// BN finalize: fold mean/var/gamma/beta into per-column scale & shift.
// ---------------------------------------------------------------------------
__global__ void bn_finalize(const float* __restrict__ colsum,
                            const float* __restrict__ colsumsq,
                            const float* __restrict__ gamma,
                            const float* __restrict__ beta,
                            float* __restrict__ scale,
                            float* __restrict__ shift,
                            float invN)
{
    int o = threadIdx.x;
    if (o < FD) {
        float mu  = colsum[o] * invN;
        float var = colsumsq[o] * invN - mu * mu;
        float inv = rsqrtf(var + BN_EPS);
        float sc  = gamma[o] * inv;
        scale[o] = sc;
        shift[o] = beta[o] - mu * sc;
    }
}

// ---------------------------------------------------------------------------
// BN apply + ReLU in place on h (== d_out), float4-vectorized (96 % 4 == 0).
// ---------------------------------------------------------------------------
__global__ __launch_bounds__(256) void bn_apply_relu(
    float* __restrict__ h,
    const float* __restrict__ scale,
    const float* __restrict__ shift,
    int n4)
{
    int i = blockIdx.x * blockDim.x + threadIdx.x;
    if (i >= n4) return;
    float4 v = ((float4*)h)[i];
    int c4 = (i % (FD / 4)) * 4;
    v.x = fmaxf(v.x * scale[c4 + 0] + shift[c4 + 0], 0.0f);
    v.y = fmaxf(v.y * scale[c4 + 1] + shift[c4 + 1], 0.0f);
    v.z = fmaxf(v.z * scale[c4 + 2] + shift[c4 + 2], 0.0f);
    v.w = fmaxf(v.w * scale[c4 + 3] + shift[c4 + 3], 0.0f);
    ((float4*)h)[i] = v;
}

// ---------------------------------------------------------------------------
// Host-side launcher (graph-capture safe: only async kernel launches).
// ---------------------------------------------------------------------------
extern "C" void kernel_launch(void* const* d_in, const int* in_sizes, int n_in,
                              void* d_out, int out_size, void* d_ws, size_t ws_size,
                              hipStream_t stream)
{
    const float* x     = (const float*)d_in[0];
    const int*   ei    = (const int*)  d_in[1];   // edge_index as int32 per contract
    const float* Wl    = (const float*)d_in[2];
    const float* bl    = (const float*)d_in[3];
    const float* Wr    = (const float*)d_in[4];
    const float* gamma = (const float*)d_in[5];
    const float* beta  = (const float*)d_in[6];

    const int N = in_sizes[0] / FD;     // 50000
    const int E = in_sizes[1] / 2;      // 800000

    // Workspace layout (floats): msg[N*96] | deg[N] | colsum[96] | colsumsq[96]
    //                            | scale[96] | shift[96]
    float* msg      = (float*)d_ws;
    float* deg      = msg + (size_t)N * FD;
    float* colsum   = deg + N;
    float* colsumsq = colsum + FD;
    float* scale    = colsumsq + FD;
    float* shift    = scale + FD;

    float* h = (float*)d_out;           // reuse output buffer as h scratch

    // 1) zero accumulators (msg, deg, colsum, colsumsq are contiguous)
    int nz = N * FD + N + 2 * FD;
    zero_f32<<<(nz + 255) / 256, 256, 0, stream>>>(msg, nz);

    // 2) edge scatter (wave per edge)
    long long tot = (long long)E * 32;
    sage_scatter<<<(int)((tot + 255) / 256), 256, 0, stream>>>(x, ei, msg, deg, E);

    // 3) fused dual GEMM via V_WMMA_F32_16X16X4_F32 + BN partial sums
    sage_gemm_wmma<<<N / 16, 192, 0, stream>>>(x, msg, deg, Wl, bl, Wr,
                                               h, colsum, colsumsq);

    // 4) BN finalize (tiny)
    bn_finalize<<<1, 128, 0, stream>>>(colsum, colsumsq, gamma, beta,
                                       scale, shift, 1.0f / (float)N);

    // 5) BN apply + ReLU, in place on d_out
    int n4 = N * FD / 4;
    bn_apply_relu<<<(n4 + 255) / 256, 256, 0, stream>>>(h, scale, shift, n4);
}